// SepnetLinear_QKV_31636729102794
// MI455X (gfx1250) — compile-verified
//
#include <hip/hip_runtime.h>

// SepnetLinear_QKV on gfx1250: out[b,n] = A @ x[b,n]^T @ B
// x: [8,197,196,768] f32 | A: [64,768] f32 | B: [196,64] f32 | out: [8,197,64,64] f32
// Uses V_WMMA_F32_16X16X4_F32 (wave32 WMMA), two-stage GEMM through LDS.

typedef float v2f __attribute__((ext_vector_type(2)));
typedef float v8f __attribute__((ext_vector_type(8)));

#define BATCH   8
#define NTOK    197
#define PDIM    196
#define CDIM    768
#define AW      64
#define BH      64
#define PTILES  13      // ceil(196/16)
#define TSTRIDE 212     // padded LDS row stride in floats (bank-conflict avoidance)

__global__ __launch_bounds__(256)
void sepnet_qkv_wmma_f32(const float* __restrict__ x,
                         const float* __restrict__ A,
                         const float* __restrict__ B,
                         float* __restrict__ out)
{
    __shared__ float tempS[AW * TSTRIDE];   // temp[64][212] = A @ x^T (padded)

    const int bn   = blockIdx.x;            // (b,n) flat, 0..1575
    const int tid  = threadIdx.x;
    const int wave = tid >> 5;              // 0..7 (wave32)
    const int lane = tid & 31;
    const int lo   = lane & 15;             // M / N index within tile
    const int hi   = lane >> 4;             // selects K half (k = 2*hi + v)

    const float* __restrict__ xb = x + (size_t)bn * (PDIM * CDIM);

    // ---------------- Phase 1: temp[64 x 196] = A @ x^T, K = 768 ----------------
    for (int pt = wave; pt < PTILES; pt += 8) {
        const int p0   = pt * 16;
        int       prow = p0 + lo;               // B-fragment N-index -> x patch row
        // Clamp instead of masking: pad columns (p >= 196) of temp are never
        // read in phase 2 (kk <= 195), so duplicated row data is harmless and
        // the inner loop stays branch-free with full-EXEC loads.
        if (prow > PDIM - 1) prow = PDIM - 1;
        const float* __restrict__ xrow = xb + prow * CDIM;

        // hint the streaming read of this x tile (global_prefetch_b8)
        __builtin_prefetch(xrow, 0, 1);

        v8f acc0 = {0.f,0.f,0.f,0.f,0.f,0.f,0.f,0.f};
        v8f acc1 = acc0, acc2 = acc0, acc3 = acc0;

        #pragma unroll 4
        for (int k = 0; k < CDIM; k += 4) {
            const int kk = k + 2 * hi;
            // B-fragment: x^T[4 x 16] -> contiguous float2 per lane
            v2f xf = *(const v2f*)(xrow + kk);
            // A-fragments for the four 16-row blocks of A (L2 resident)
            v2f a0 = *(const v2f*)(A + (0*16 + lo) * CDIM + kk);
            v2f a1 = *(const v2f*)(A + (1*16 + lo) * CDIM + kk);
            v2f a2 = *(const v2f*)(A + (2*16 + lo) * CDIM + kk);
            v2f a3 = *(const v2f*)(A + (3*16 + lo) * CDIM + kk);

            acc0 = __builtin_amdgcn_wmma_f32_16x16x4_f32(false, a0, false, xf, (short)0, acc0, false, false);
            acc1 = __builtin_amdgcn_wmma_f32_16x16x4_f32(false, a1, false, xf, (short)0, acc1, false, false);
            acc2 = __builtin_amdgcn_wmma_f32_16x16x4_f32(false, a2, false, xf, (short)0, acc2, false, false);
            acc3 = __builtin_amdgcn_wmma_f32_16x16x4_f32(false, a3, false, xf, (short)0, acc3, false, false);
        }

        // C/D layout: d[v] = D[m = v + 8*hi][n = lo]  ->  tempS[m][p0+lo]
        #pragma unroll
        for (int v = 0; v < 8; ++v) {
            const int m = v + 8 * hi;
            tempS[(0*16 + m) * TSTRIDE + p0 + lo] = acc0[v];
            tempS[(1*16 + m) * TSTRIDE + p0 + lo] = acc1[v];
            tempS[(2*16 + m) * TSTRIDE + p0 + lo] = acc2[v];
            tempS[(3*16 + m) * TSTRIDE + p0 + lo] = acc3[v];
        }
    }

    __syncthreads();

    // ---------------- Phase 2: out[64 x 64] = temp @ B, K = 196 ----------------
    float* __restrict__ op = out + (size_t)bn * (AW * BH);

    for (int t = wave; t < 16; t += 8) {        // 4x4 output tiles, 2 per wave
        const int m0 = (t >> 2) * 16;
        const int h0 = (t & 3) * 16;

        v8f acc = {0.f,0.f,0.f,0.f,0.f,0.f,0.f,0.f};

        #pragma unroll 7
        for (int p = 0; p < PDIM; p += 4) {     // 49 exact K-steps (7x7)
            const int kk = p + 2 * hi;
            // A-fragment from LDS: temp[m0+lo][kk .. kk+1] (8B-aligned ds_load_b64)
            v2f af = *(const v2f*)(&tempS[(m0 + lo) * TSTRIDE + kk]);
            // B-fragment: B[kk][h0+lo], B[kk+1][h0+lo] (row stride 64)
            v2f bf;
            bf.x = B[(kk + 0) * BH + h0 + lo];
            bf.y = B[(kk + 1) * BH + h0 + lo];
            acc = __builtin_amdgcn_wmma_f32_16x16x4_f32(false, af, false, bf, (short)0, acc, false, false);
        }

        // Streaming output: non-temporal stores (no reuse of out lines)
        #pragma unroll
        for (int v = 0; v < 8; ++v)
            __builtin_nontemporal_store(acc[v], &op[(m0 + v + 8 * hi) * BH + h0 + lo]);
    }
}

extern "C" void kernel_launch(void* const* d_in, const int* in_sizes, int n_in,
                              void* d_out, int out_size, void* d_ws, size_t ws_size,
                              hipStream_t stream) {
    const float* x = (const float*)d_in[0];   // [8,197,196,768]
    const float* A = (const float*)d_in[1];   // [64,768]
    const float* B = (const float*)d_in[2];   // [196,64]
    float* out = (float*)d_out;               // [8,197,64,64]

    dim3 grid(BATCH * NTOK);                  // 1576 workgroups, one per (b,n)
    dim3 block(256);                          // 8 wave32 waves
    sepnet_qkv_wmma_f32<<<grid, block, 0, stream>>>(x, A, B, out);
}